// TFAdaptiveAvgPool2D_7430293422930
// MI455X (gfx1250) — compile-verified
//
#include <hip/hip_runtime.h>
#include <stdint.h>

// ---------------------------------------------------------------------------
// Adaptive avg pool (32,50,50,768) -> (32,7,7,768), NHWC, f32.
// 50 -> 7 gives uniform 8-wide windows at starts 7*i on both axes, so each
// output = sum(8x8 window) / 64 (exact: /8 then /8 == /64 in fp).
// HBM-bound (AI ~ 0.3 flop/B, ideal ~10.8us at 23.3 TB/s): stream the input
// via the CDNA5 Tensor Data Mover into LDS (two 3-D tile descriptors,
// double-buffered on TENSORcnt), reduce with wave32 VALU, one b128 store.
// 3-D grid avoids all div/mod scalar index math before the first TDM issue.
// ---------------------------------------------------------------------------

#define N_B   32
#define IH    50
#define IW    50
#define C_    768
#define OH    7
#define OW    7
#define KH    8
#define KW    8
#define CHALF 384                       // channels handled per block
#define HROWS 4                         // H rows per TDM tile (2 tiles = 8)
#define BUF_FLOATS (HROWS * KW * CHALF) // 12288 floats
#define BUF_BYTES  (BUF_FLOATS * 4)     // 49152 bytes per buffer
#define NTHR  96                        // 96 threads * float4 = 384 channels

typedef uint32_t u32x4 __attribute__((ext_vector_type(4)));
typedef int32_t  i32x4 __attribute__((ext_vector_type(4)));
typedef int32_t  i32x8 __attribute__((ext_vector_type(8)));

#ifndef __has_builtin
#define __has_builtin(x) 0
#endif

#if defined(__AMDGCN__) && __has_builtin(__builtin_amdgcn_tensor_load_to_lds)
#define HAVE_TDM_DEVICE 1
#else
#define HAVE_TDM_DEVICE 0
#endif

#if __has_include(<hip/amd_detail/amd_gfx1250_TDM.h>)
#define TDM_SIX_ARG 1   // therock-10.0 headers => 6-arg builtin
#else
#define TDM_SIX_ARG 0   // ROCm 7.2 => 5-arg builtin
#endif

__global__ __launch_bounds__(NTHR) void adaptive_pool_7x7_tdm(
    const float* __restrict__ in, float* __restrict__ out) {
  extern __shared__ float smem[];   // 2 * BUF_BYTES dynamic LDS

  // 3-D grid: x = (ow, chalf), y = oh, z = n  -> no div/mod decode.
  const int ch = blockIdx.x & 1;
  const int ow = blockIdx.x >> 1;
  const int oh = blockIdx.y;
  const int n  = blockIdx.z;

  const int hs = 7 * oh;            // floor(i*50/7) == 7*i for all i in [0,7)
  const int ws = 7 * ow;
  const int c0 = ch * CHALF;
  const int t  = threadIdx.x;

  float ax = 0.f, ay = 0.f, az = 0.f, aw = 0.f;

#if HAVE_TDM_DEVICE
  // ---- wave 0 issues two 3-D TDM tile loads (4 H rows x 8 W x 384 C each) ----
  if (t < 32) {
    const uint32_t lds_base = (uint32_t)(uintptr_t)(void*)smem; // flat low32 = LDS offset
    #pragma unroll
    for (int buf = 0; buf < 2; ++buf) {
      const uint64_t ga = (uint64_t)(uintptr_t)in +
          4ull * ((((uint64_t)(n * IH + hs + HROWS * buf)) * IW + ws) * C_ + c0);
      // D# group 0: count=1, lds_addr, global_addr[56:0], type=2
      u32x4 g0 = { 1u,
                   lds_base + (uint32_t)(buf * BUF_BYTES),
                   (uint32_t)ga,
                   (uint32_t)((ga >> 32) & 0x01FFFFFFu) | (2u << 30) };
      // D# group 1:
      //  [17:16] data_size=2 (4B)          [79:48]  tensor_dim0 = 768
      //  [111:80] tensor_dim1 = 50         [127:112] tile_dim0 = 384
      //  [143:128] tile_dim1 = 8 (W)       [159:144] tile_dim2 = 4 (H)
      //  [207:160] tensor_dim0_stride=768  [255:208] tensor_dim1_stride=38400
      i32x8 g1 = { (int)(2u << 16),
                   (int)(768u << 16),
                   (int)(50u << 16),
                   (int)(384u << 16),
                   (int)((4u << 16) | 8u),
                   (int)768,
                   (int)(38400u << 16),
                   0 };
      // D# group 2: tensor_dim2 = 50; tile_dim3 = 0 (3-D tile only)
      i32x4 g2 = { 50, 0, 0, 0 };
      i32x4 g3 = { 0, 0, 0, 0 };
#if TDM_SIX_ARG
      i32x8 g4 = { 0, 0, 0, 0, 0, 0, 0, 0 };
      __builtin_amdgcn_tensor_load_to_lds(g0, g1, g2, g3, g4, 0);
#else
      __builtin_amdgcn_tensor_load_to_lds(g0, g1, g2, g3, 0);
#endif
    }
    // First tile complete (tensor ops from one wave complete in order).
    __builtin_amdgcn_s_wait_tensorcnt(1);
  }
  __syncthreads();

  // ---- reduce buffer 0 while buffer 1 DMA is still in flight ----
  {
    const float4* s0 = (const float4*)smem;
    #pragma unroll
    for (int p = 0; p < HROWS * KW; ++p) {
      float4 v = s0[p * (CHALF / 4) + t];
      ax += v.x; ay += v.y; az += v.z; aw += v.w;
    }
  }
  if (t < 32) __builtin_amdgcn_s_wait_tensorcnt(0);
  __syncthreads();

  // ---- reduce buffer 1 ----
  {
    const float4* s1 = (const float4*)(smem + BUF_FLOATS);
    #pragma unroll
    for (int p = 0; p < HROWS * KW; ++p) {
      float4 v = s1[p * (CHALF / 4) + t];
      ax += v.x; ay += v.y; az += v.z; aw += v.w;
    }
  }
#else
  // Fallback (no TDM builtin): direct coalesced float4 streaming reads.
  const float4* gp = (const float4*)in;
  #pragma unroll
  for (int dh = 0; dh < KH; ++dh) {
    #pragma unroll
    for (int dw = 0; dw < KW; ++dw) {
      const size_t base =
          (((size_t)(n * IH + hs + dh)) * IW + (ws + dw)) * C_ + c0;
      float4 v = gp[base / 4 + t];
      ax += v.x; ay += v.y; az += v.z; aw += v.w;
    }
  }
#endif

  const float s = 1.0f / 64.0f;       // exact: matches reference /8 then /8
  float4 r;
  r.x = ax * s; r.y = ay * s; r.z = az * s; r.w = aw * s;
  const size_t oidx = (((size_t)(n * OH + oh)) * OW + ow) * C_ + c0 + 4 * t;
  *(float4*)&out[oidx] = r;
}

extern "C" void kernel_launch(void* const* d_in, const int* in_sizes, int n_in,
                              void* d_out, int out_size, void* d_ws, size_t ws_size,
                              hipStream_t stream) {
  (void)in_sizes; (void)n_in; (void)out_size; (void)d_ws; (void)ws_size;
  const float* in = (const float*)d_in[0];
  float* out = (float*)d_out;

  dim3 grid(OW * 2, OH, N_B);     // (ow,chalf) x oh x n  -> 3136 blocks
  dim3 block(NTHR);               // 3 waves of wave32
  adaptive_pool_7x7_tdm<<<grid, block, 2 * BUF_BYTES, stream>>>(in, out);
}